// Detect_50431505989817
// MI455X (gfx1250) — compile-verified
//
#include <hip/hip_runtime.h>
#include <stdint.h>

typedef float v2f __attribute__((ext_vector_type(2)));
typedef float v8f __attribute__((ext_vector_type(8)));

#define BM 128
#define BN 64
#define KC 16
#define LDA 18   // A row stride (floats): even -> 8B-aligned b64 frag reads
#define LDB 20   // B row stride (floats): 80B rows -> 16B-aligned b128 async stores,
                 // and 20*n mod 64 distinct for n=0..15 -> conflict-free frag reads

__device__ __forceinline__ float fast_sigmoid(float t) {
    return __builtin_amdgcn_rcpf(1.0f + __expf(-t));
}

// One detection level. LEVEL=0: K=256,G=64,s=8 ; LEVEL=1: K=512,G=32,s=16
template <int LEVEL>
__global__ __launch_bounds__(256, 2)
void detect_head_kernel(const float* __restrict__ feat,
                        const float* __restrict__ Wt,
                        const float* __restrict__ bias,
                        float* __restrict__ out)
{
    constexpr int   K        = (LEVEL == 0) ? 256 : 512;
    constexpr int   G        = (LEVEL == 0) ? 64  : 32;
    constexpr int   GG       = G * G;
    constexpr int   N        = 1548;               // 18 anchors * 86 ch
    constexpr int   ROWS_TOT = 92160;              // 18*(64*64 + 32*32)
    constexpr int   LVLBASE  = (LEVEL == 0) ? 0 : 18 * 64 * 64;
    constexpr float SXYV     = (LEVEL == 0) ? 1.2f : 1.1f;
    constexpr float SV       = (LEVEL == 0) ? 8.0f : 16.0f;
    constexpr float OFFC     = -(SXYV - 1.0f) * 0.5f;
    constexpr float AW0 = (LEVEL == 0) ? 10.f : 30.f;
    constexpr float AW1 = (LEVEL == 0) ? 16.f : 62.f;
    constexpr float AW2 = (LEVEL == 0) ? 33.f : 59.f;
    constexpr float AH0 = (LEVEL == 0) ? 13.f : 61.f;
    constexpr float AH1 = (LEVEL == 0) ? 30.f : 45.f;
    constexpr float AH2 = (LEVEL == 0) ? 23.f : 119.f;

    __shared__ float As[2][BM * LDA];
    __shared__ float Bs[2][BN * LDB];

    const int t    = threadIdx.x;
    const int wave = t >> 5;
    const int lane = t & 31;
    const int half = lane >> 4;   // K-half selector in 32-bit fragment layout
    const int l16  = lane & 15;

    const int m0 = blockIdx.x * BM;          // rows = b*GG + pos
    const int n0 = blockIdx.y * BN;          // cols = a*86 + c

    const int b    = m0 / GG;                // whole block shares one batch image
    const int pos0 = m0 - b * GG;            // 128 contiguous spatial positions

    const float* fbase = feat + (size_t)b * K * GG + pos0;   // A[m][k] = fbase[k*GG + m]

    // wave grid 4(m) x 2(n); each wave owns a 32x32 C tile = 2x2 fragments
    const int wm = (wave & 3) << 5;
    const int wn = (wave >> 2) << 5;

    v8f acc00 = {0.f, 0.f, 0.f, 0.f, 0.f, 0.f, 0.f, 0.f};
    v8f acc01 = acc00, acc10 = acc00, acc11 = acc00;

    // ---- async global->LDS staging mapping (256 threads) ----
    const int a_mi = t & 127;                // A: row 0..127 (coalesced across lanes)
    const int a_k8 = (t >> 7) << 3;          // A: k base 0 or 8
    const int b_n  = t >> 2;                 // B: output column row 0..63
    const int b_kq = (t & 3) << 2;           // B: k offset 0,4,8,12
    const int bn_c = min(n0 + b_n, N - 1);   // clamp: no OOB, bad cols never stored

    // LDS byte addresses: generic shared ptr truncated -> raw LDS offset
    const uint32_t asBase = (uint32_t)(uintptr_t)&As[0][0];
    const uint32_t bsBase = (uint32_t)(uintptr_t)&Bs[0][0];
    constexpr uint32_t ABYTES = (uint32_t)(BM * LDA * 4);
    constexpr uint32_t BBYTES = (uint32_t)(BN * LDB * 4);

    const uint32_t ldsA0 = asBase + (uint32_t)(a_mi * LDA + a_k8) * 4u;
    const uint32_t ldsB0 = bsBase + (uint32_t)(b_n * LDB + b_kq) * 4u;
    const uint32_t voffB = (uint32_t)(bn_c * K + b_kq) * 4u;   // per-lane, stage-invariant

    auto issue_stage = [&](int bufsel, int k0) {
        const float* sA = fbase + (size_t)k0 * GG;   // uniform -> SGPR pair
        const float* sW = Wt + k0;                   // uniform -> SGPR pair
        const uint32_t lA = ldsA0 + (uint32_t)bufsel * ABYTES;
        const uint32_t lB = ldsB0 + (uint32_t)bufsel * BBYTES;
        #pragma unroll
        for (int i = 0; i < 8; ++i) {
            const uint32_t vA = (uint32_t)((a_k8 + i) * GG + a_mi) * 4u;
            asm volatile("global_load_async_to_lds_b32 %0, %1, %2"
                         :: "v"(lA + 4u * (uint32_t)i), "v"(vA), "s"(sA)
                         : "memory");
        }
        asm volatile("global_load_async_to_lds_b128 %0, %1, %2"
                     :: "v"(lB), "v"(voffB), "s"(sW)
                     : "memory");
    };

    // ---- software-pipelined K loop: 1 barrier per stage, async double buffer ----
    issue_stage(0, 0);
    int buf = 0;
    for (int k0 = 0; k0 < K; k0 += KC) {
        asm volatile("s_wait_asynccnt 0x0" ::: "memory");  // own writes to buf done
        __syncthreads();                                   // all waves' writes visible;
                                                           // all reads of buf^1 retired
        if (k0 + KC < K) issue_stage(buf ^ 1, k0 + KC);    // overlap with WMMA below

        const float* Ab = &As[buf][0];
        const float* Bb = &Bs[buf][0];
        // ISA 32-bit 16x4 fragment: lane L, vgpr v <-> (row L%16, k = 2*(L/16)+v)
        #pragma unroll
        for (int ks = 0; ks < KC; ks += 4) {
            const int kf = ks + (half << 1);
            const v2f af0 = *(const v2f*)&Ab[(wm + l16) * LDA + kf];
            const v2f af1 = *(const v2f*)&Ab[(wm + 16 + l16) * LDA + kf];
            const v2f bf0 = *(const v2f*)&Bb[(wn + l16) * LDB + kf];
            const v2f bf1 = *(const v2f*)&Bb[(wn + 16 + l16) * LDB + kf];
            acc00 = __builtin_amdgcn_wmma_f32_16x16x4_f32(false, af0, false, bf0,
                                                          (short)0, acc00, false, false);
            acc01 = __builtin_amdgcn_wmma_f32_16x16x4_f32(false, af0, false, bf1,
                                                          (short)0, acc01, false, false);
            acc10 = __builtin_amdgcn_wmma_f32_16x16x4_f32(false, af1, false, bf0,
                                                          (short)0, acc10, false, false);
            acc11 = __builtin_amdgcn_wmma_f32_16x16x4_f32(false, af1, false, bf1,
                                                          (short)0, acc11, false, false);
        }
        buf ^= 1;
    }

    // ---- fused decode epilogue + scatter into [B, 92160, 86] ----
    #pragma unroll
    for (int f = 0; f < 4; ++f) {
        const v8f acc = (f == 0) ? acc00 : (f == 1) ? acc01 : (f == 2) ? acc10 : acc11;
        const int nf = f & 1;
        const int mf = f >> 1;
        const int o  = n0 + wn + (nf << 4) + l16;      // output channel column
        const int pos_wave = pos0 + wm + (mf << 4);    // + (v + 8*half) = spatial pos
        if (o < N) {
            const int a     = o / 86;
            const int c     = o - a * 86;
            const int shape = a / 6;
            const int ang   = a - shape * 6;
            const float aw  = (shape == 0) ? AW0 : ((shape == 1) ? AW1 : AW2);
            const float ah  = (shape == 0) ? AH0 : ((shape == 1) ? AH1 : AH2);
            const float angle = (ang == 5) ? 1.5707963f
                                           : (float)(ang - 2) * 0.5235988f;
            const float bia = bias[o];
            float* obase = out + ((size_t)b * ROWS_TOT + LVLBASE + (size_t)a * GG) * 86 + c;
            #pragma unroll
            for (int v = 0; v < 8; ++v) {
                const int pos = pos_wave + (half << 3) + v;   // C layout: M = v + 8*half
                const int yy  = pos / G;
                const int xx  = pos & (G - 1);
                const float tv = acc[v] + bia;
                const float sg = fast_sigmoid(tv);
                const float ex = __expf(tv);
                float r = sg;                                   // conf / cls default
                if (c == 0)      r = (sg * SXYV + OFFC + (float)xx) * SV;
                else if (c == 1) r = (sg * SXYV + OFFC + (float)yy) * SV;
                else if (c == 2) r = ex * aw;
                else if (c == 3) r = ex * ah;
                else if (c == 4) r = tv + angle;
                obase[(size_t)pos * 86] = r;
            }
        }
    }
}

extern "C" void kernel_launch(void* const* d_in, const int* in_sizes, int n_in,
                              void* d_out, int out_size, void* d_ws, size_t ws_size,
                              hipStream_t stream) {
    (void)in_sizes; (void)n_in; (void)out_size; (void)d_ws; (void)ws_size;
    const float* x0 = (const float*)d_in[0];
    const float* x1 = (const float*)d_in[1];
    const float* W0 = (const float*)d_in[2];
    const float* b0 = (const float*)d_in[3];
    const float* W1 = (const float*)d_in[4];
    const float* b1 = (const float*)d_in[5];
    float* out = (float*)d_out;

    dim3 blk(256);
    dim3 g0(8 * 64 * 64 / BM, (1548 + BN - 1) / BN);   // 256 x 25
    dim3 g1(8 * 32 * 32 / BM, (1548 + BN - 1) / BN);   // 64 x 25
    detect_head_kernel<0><<<g0, blk, 0, stream>>>(x0, W0, b0, out);
    detect_head_kernel<1><<<g1, blk, 0, stream>>>(x1, W1, b1, out);
}